// L2NonDeblurCG_40870908788887
// MI455X (gfx1250) — compile-verified
//
#include <hip/hip_runtime.h>

// ---------------- problem geometry ----------------
#define BH 384
#define BW 384
#define NB (3*BH*BW)          // blur elements (and output)
#define Hh 408                // padded image H (384 + 2*12)
#define Ww 408
#define HWp (Hh*Ww)
#define IMG (3*HWp)
#define KS 25                 // PSF size
#define PS 12                 // KS/2

typedef __attribute__((ext_vector_type(2))) float v2f;
typedef __attribute__((ext_vector_type(8))) float v8f;

__device__ __forceinline__ int wrapi(int v, int n) { int m = v % n; return (m < 0) ? m + n : m; }
// interior mask of the padded image (mask = 1 on [12,396) x [12,396))
__device__ __forceinline__ float maskf(int r, int c) {
    return (r >= PS && r < Hh - PS && c >= PS && c < Ww - PS) ? 1.0f : 0.0f;
}
// circular 5-point operator: dual_conv(v,G1,ones)+dual_conv(v,G2,ones) == 4x - up - dn - lf - rt
__device__ __forceinline__ float lap_at(const float* __restrict__ v, int ch, int r, int c) {
    const float* p = v + ch * HWp;
    float x  = p[r * Ww + c];
    float up = p[wrapi(r - 1, Hh) * Ww + c];
    float dn = p[wrapi(r + 1, Hh) * Ww + c];
    float lf = p[r * Ww + wrapi(c - 1, Ww)];
    float rt = p[r * Ww + wrapi(c + 1, Ww)];
    return 4.0f * x - up - dn - lf - rt;
}

// ---------------- flip 25x25 kernel (both dims == reverse flat) ----------------
__global__ void flipk_kernel(const float* __restrict__ k, float* __restrict__ kf) {
    int i = blockIdx.x * blockDim.x + threadIdx.x;
    if (i < KS * KS) kf[i] = k[KS * KS - 1 - i];
}

// ---------------- median_pool5 (reflect pad 2) + noise = blur - median ----------------
__global__ void median_noise_kernel(const float* __restrict__ blur, float* __restrict__ noise) {
    int idx = blockIdx.x * blockDim.x + threadIdx.x;
    if (idx >= NB) return;
    int c  = idx % BW;
    int r  = (idx / BW) % BH;
    int ch = idx / (BH * BW);
    const float* src = blur + ch * BH * BW;
    float v[25];
    int t = 0;
    for (int dr = -2; dr <= 2; ++dr) {
        int rr = r + dr;
        if (rr < 0) rr = -rr;
        if (rr >= BH) rr = 2 * BH - 2 - rr;
        for (int dc = -2; dc <= 2; ++dc) {
            int cc = c + dc;
            if (cc < 0) cc = -cc;
            if (cc >= BW) cc = 2 * BW - 2 - cc;
            v[t++] = src[rr * BW + cc];
        }
    }
    // insertion sort 25; median = v[12]
    for (int a = 1; a < 25; ++a) {
        float key = v[a];
        int b = a - 1;
        while (b >= 0 && v[b] > key) { v[b + 1] = v[b]; --b; }
        v[b + 1] = key;
    }
    noise[idx] = blur[idx] - v[12];
}

// ---------------- two-stage stats reduction -> sigma ----------------
__global__ void stats1_kernel(const float* __restrict__ noise, float* __restrict__ part, int n) {
    __shared__ float s1[256];
    __shared__ float s2[256];
    float a = 0.0f, b = 0.0f;
    for (int i = blockIdx.x * blockDim.x + threadIdx.x; i < n; i += gridDim.x * blockDim.x) {
        float v = noise[i];
        a += v; b += v * v;
    }
    s1[threadIdx.x] = a; s2[threadIdx.x] = b;
    __syncthreads();
    for (int s = 128; s > 0; s >>= 1) {
        if (threadIdx.x < s) { s1[threadIdx.x] += s1[threadIdx.x + s]; s2[threadIdx.x] += s2[threadIdx.x + s]; }
        __syncthreads();
    }
    if (threadIdx.x == 0) { part[2 * blockIdx.x] = s1[0]; part[2 * blockIdx.x + 1] = s2[0]; }
}

__global__ void stats2_kernel(const float* __restrict__ part, int nblocks, float* __restrict__ scal) {
    __shared__ float s1[256];
    __shared__ float s2[256];
    float a = 0.0f, b = 0.0f;
    for (int i = threadIdx.x; i < nblocks; i += blockDim.x) { a += part[2 * i]; b += part[2 * i + 1]; }
    s1[threadIdx.x] = a; s2[threadIdx.x] = b;
    __syncthreads();
    for (int s = 128; s > 0; s >>= 1) {
        if (threadIdx.x < s) { s1[threadIdx.x] += s1[threadIdx.x + s]; s2[threadIdx.x] += s2[threadIdx.x + s]; }
        __syncthreads();
    }
    if (threadIdx.x == 0) {
        float N = (float)NB;
        float var = (s2[0] - s1[0] * s1[0] / N) / (N - 1.0f);
        scal[0] = s1[0];
        scal[1] = s2[0];
        scal[2] = sqrtf(var > 0.0f ? var : 0.0f);   // sigma
    }
}

// ---------------- edge pad blur (3,384,384) -> x (3,408,408) ----------------
__global__ void pad_edge_kernel(const float* __restrict__ blur, float* __restrict__ xp) {
    int idx = blockIdx.x * blockDim.x + threadIdx.x;
    if (idx >= IMG) return;
    int c  = idx % Ww;
    int r  = (idx / Ww) % Hh;
    int ch = idx / HWp;
    int rr = r - PS; rr = rr < 0 ? 0 : (rr > BH - 1 ? BH - 1 : rr);
    int cc = c - PS; cc = cc < 0 ? 0 : (cc > BW - 1 ? BW - 1 : cc);
    xp[idx] = blur[ch * BH * BW + rr * BW + cc];
}

// ---------------- 25x25 circular convolution as banded WMMA GEMM ----------------
// out(r,c) = sum_{i,j} k(i,j) * x((r+i-12) mod H, (c+j-12) mod W)
// Per 16x16 tile: for each kernel row i, banded GEMM with K=40:
//   out(m,n) += sum_u X(m+i, u) * B_i(u,n),   B_i(u,n) = k(i, u-n) for 0<=u-n<25
// Kernel rows are staged ZERO-PADDED (64-wide, +16 left guard) so B fetches are
// unconditional LDS loads — no divergent predication around the WMMAs.
__global__ __launch_bounds__(32)
void conv_circ_wmma(const float* __restrict__ src, const float* __restrict__ kern,
                    float* __restrict__ dst, int premask, int postmask,
                    const float* __restrict__ lsrc, const float* __restrict__ scal, int addlap) {
    __shared__ float xs[40 * 41];    // 40x40 input patch (stride 41)
    __shared__ float ksp[KS * 64];   // zero-padded kernel rows: ksp[i*64 + 16 + j] = k(i,j)
    const int ch = blockIdx.z;
    const int r0 = blockIdx.y * 16;
    const int c0 = blockIdx.x * 16;
    const int lane = threadIdx.x;
    const float* s = src + ch * HWp;

    // stage input patch (rows r0-12..r0+27, cols c0-12..c0+27), circular wrap, optional premask
    for (int idx = lane; idx < 1600; idx += 32) {
        int rr = idx / 40, cc = idx - rr * 40;
        int sr = wrapi(r0 - PS + rr, Hh);
        int sc = wrapi(c0 - PS + cc, Ww);
        float v = s[sr * Ww + sc];
        if (premask) v *= maskf(sr, sc);
        xs[rr * 41 + cc] = v;
    }
    // stage zero-padded kernel rows
    for (int idx = lane; idx < KS * 64; idx += 32) ksp[idx] = 0.0f;
    __syncthreads();
    for (int idx = lane; idx < KS * KS; idx += 32) {
        int i = idx / KS, j = idx - i * KS;
        ksp[i * 64 + 16 + j] = kern[idx];
    }
    __syncthreads();

    const int m = lane & 15;                 // A row index (M) and B col index (N)
    const int khalf = (lane >> 4) << 1;      // lanes 16-31 hold K=2,3 of each K-quad
    v8f acc = {};
    #pragma unroll 5
    for (int i = 0; i < 25; ++i) {
        const float* xrow = &xs[(m + i) * 41];
        const float* krow = &ksp[i * 64 + 16 - m];   // index by u0 directly, always in-bounds
        #pragma unroll
        for (int kb = 0; kb < 10; ++kb) {
            int u0 = kb * 4 + khalf;
            v2f a, b;
            a.x = xrow[u0];
            a.y = xrow[u0 + 1];
            b.x = krow[u0];
            b.y = krow[u0 + 1];
            acc = __builtin_amdgcn_wmma_f32_16x16x4_f32(false, a, false, b,
                                                        (short)0, acc, false, false);
        }
    }

    // C/D layout: lane l: N = l&15; VGPR v holds M = v + (l>=16 ? 8 : 0)
    float* d = dst + ch * HWp;
    const int col = c0 + (lane & 15);
    const int mo = (lane >> 4) * 8;
    const float sg = addlap ? scal[2] : 0.0f;
    if (col < Ww) {
        #pragma unroll
        for (int v = 0; v < 8; ++v) {
            int row = r0 + mo + v;
            if (row < Hh) {
                float val = acc[v];
                if (postmask) val *= maskf(row, col);
                if (addlap)   val += sg * lap_at(lsrc, ch, row, col);
                d[row * Ww + col] = val;
            }
        }
    }
}

// ---------------- residual init: r = b - A(x)  (q already holds A(x)) ----------------
__global__ void residual_kernel(float* __restrict__ r, const float* __restrict__ q) {
    int idx = blockIdx.x * blockDim.x + threadIdx.x;
    if (idx >= IMG) return;
    r[idx] -= q[idx];
}

// ---------------- per-channel dot product (one block per channel) ----------------
__global__ void dot3_kernel(const float* __restrict__ a, const float* __restrict__ b,
                            float* __restrict__ out) {
    __shared__ float sd[256];
    int ch = blockIdx.x;
    const float* pa = a + ch * HWp;
    const float* pb = b + ch * HWp;
    float s = 0.0f;
    for (int i = threadIdx.x; i < HWp; i += blockDim.x) s += pa[i] * pb[i];
    sd[threadIdx.x] = s;
    __syncthreads();
    for (int st = 128; st > 0; st >>= 1) {
        if (threadIdx.x < st) sd[threadIdx.x] += sd[threadIdx.x + st];
        __syncthreads();
    }
    if (threadIdx.x == 0) out[ch] = sd[0];
}

// ---------------- p = r + beta*p  (beta = rho/rho_prev; first iter p = r) ----------------
__global__ void update_p_kernel(float* __restrict__ p, const float* __restrict__ r,
                                const float* __restrict__ scal, int first) {
    int idx = blockIdx.x * blockDim.x + threadIdx.x;
    if (idx >= IMG) return;
    if (first) { p[idx] = r[idx]; return; }
    int ch = idx / HWp;
    float beta = scal[3 + ch] / scal[9 + ch];
    p[idx] = r[idx] + beta * p[idx];
}

// ---------------- x += alpha*p ; r -= alpha*q ; rho_prev = rho ----------------
__global__ void update_xr_kernel(float* __restrict__ x, float* __restrict__ r,
                                 const float* __restrict__ p, const float* __restrict__ q,
                                 float* __restrict__ scal) {
    int idx = blockIdx.x * blockDim.x + threadIdx.x;
    if (idx >= IMG) return;
    int ch = idx / HWp;
    float alpha = scal[3 + ch] / scal[6 + ch];
    x[idx] += alpha * p[idx];
    r[idx] -= alpha * q[idx];
    if (idx < 3) scal[9 + idx] = scal[3 + idx];   // rho_prev <- rho
}

// ---------------- interior extract -> output (3,384,384) ----------------
__global__ void extract_kernel(const float* __restrict__ x, float* __restrict__ out) {
    int idx = blockIdx.x * blockDim.x + threadIdx.x;
    if (idx >= NB) return;
    int c  = idx % BW;
    int r  = (idx / BW) % BH;
    int ch = idx / (BH * BW);
    out[idx] = x[ch * HWp + (r + PS) * Ww + (c + PS)];
}

// =====================================================================
extern "C" void kernel_launch(void* const* d_in, const int* in_sizes, int n_in,
                              void* d_out, int out_size, void* d_ws, size_t ws_size,
                              hipStream_t stream) {
    const float* blur = (const float*)d_in[0];
    const float* kern = (const float*)d_in[1];
    float* out = (float*)d_out;
    float* ws  = (float*)d_ws;

    // workspace layout (floats):
    // [0..15]   scalars: 0 sum, 1 sumsq, 2 sigma, 3..5 rho, 6..8 pq, 9..11 rho_prev
    // [16..640] flipped kernel (625)
    // [656..4111] stage-1 partials (2*1728 = 3456)
    // [4112..]  image buffers, each IMG floats: x, r, p, q, t1
    float* scal  = ws;
    float* kflip = ws + 16;
    float* part  = ws + 656;
    float* x  = ws + 4112;
    float* r  = x + IMG;
    float* p  = r + IMG;
    float* q  = p + IMG;
    float* t1 = q + IMG;

    const int EL = 256;
    const int gIMG = (IMG + EL - 1) / EL;
    const int gNB  = (NB + EL - 1) / EL;
    dim3 gconv((Ww + 15) / 16, (Hh + 15) / 16, 3);   // 26 x 26 x 3 tiles

    // --- setup: flipped kernel, sigma from median-filter noise ---
    flipk_kernel<<<3, EL, 0, stream>>>(kern, kflip);
    median_noise_kernel<<<gNB, EL, 0, stream>>>(blur, t1);        // t1 <- noise (reused later)
    stats1_kernel<<<NB / EL, EL, 0, stream>>>(t1, part, NB);      // 1728 partials
    stats2_kernel<<<1, EL, 0, stream>>>(part, NB / EL, scal);     // scal[2] = sigma
    pad_edge_kernel<<<gIMG, EL, 0, stream>>>(blur, x);            // x = xpad

    // --- r = b - A(x):  b = conv(mask*x, k);  A(v) = conv(mask*conv(v,kf), k) + sigma*Lap(v)
    conv_circ_wmma<<<gconv, 32, 0, stream>>>(x, kern, r, 1, 0, x, scal, 0);   // r <- b
    conv_circ_wmma<<<gconv, 32, 0, stream>>>(x, kflip, t1, 0, 1, x, scal, 0); // t1 <- mask*conv(x,kf)
    conv_circ_wmma<<<gconv, 32, 0, stream>>>(t1, kern, q, 0, 0, x, scal, 1);  // q <- A(x) (lap fused)
    residual_kernel<<<gIMG, EL, 0, stream>>>(r, q);                           // r = b - A(x)

    // --- 20 CG iterations (per-channel scalars stay on device) ---
    for (int it = 0; it < 20; ++it) {
        dot3_kernel<<<3, EL, 0, stream>>>(r, r, scal + 3);                     // rho
        update_p_kernel<<<gIMG, EL, 0, stream>>>(p, r, scal, it == 0);         // p = r + beta*p
        conv_circ_wmma<<<gconv, 32, 0, stream>>>(p, kflip, t1, 0, 1, p, scal, 0);  // t1 = mask*conv(p,kf)
        conv_circ_wmma<<<gconv, 32, 0, stream>>>(t1, kern, q, 0, 0, p, scal, 1);   // q = A(p) (lap fused)
        dot3_kernel<<<3, EL, 0, stream>>>(p, q, scal + 6);                     // pq
        update_xr_kernel<<<gIMG, EL, 0, stream>>>(x, r, p, q, scal);           // x,r update; rho_prev=rho
    }

    extract_kernel<<<gNB, EL, 0, stream>>>(x, out);
}